// Quantizer_90434831385355
// MI455X (gfx1250) — compile-verified
//
#include <hip/hip_runtime.h>

// VQ-VAE quantizer for MI455X (gfx1250).
//   x: (16, 64, 64, 64) f32   -> N=65536 pixels, C=64 dims (channel stride 4096)
//   W: (1024, 64) f32 codebook
// Pass 1: per-pixel argmin over 1024 codes using V_WMMA_F32_16X16X4_F32,
//         full codebook (-2*W, 256KB) staged in the 320KB WGP LDS.
// Pass 2: gather W[idx] -> quantized (b,c,h,w) + per-block squared-error partials.
// Pass 3: deterministic reduction -> two identical loss scalars.

#define NPIX   65536
#define NCODE  1024
#define CDIM   64
#define LDS_STRIDE 65   // pad code rows to 65 floats: breaks 64-bank conflicts

typedef __attribute__((ext_vector_type(2))) float v2f;
typedef __attribute__((ext_vector_type(8))) float v8f;

// flat (b,h,w) pixel n, channel c  ->  offset in (b,c,h,w) tensor
__device__ __forceinline__ int xoff(int n, int c) {
  return ((n >> 12) << 18) + (c << 12) + (n & 4095);
}

extern __shared__ float lds[];

__global__ __launch_bounds__(256) void vq_argmin_wmma(
    const float* __restrict__ x, const float* __restrict__ W,
    int* __restrict__ outIdx)
{
  float* ldsW  = lds;                        // [1024][65] holds -2*W
  float* wnorm = lds + NCODE * LDS_STRIDE;   // [1024]     holds |W_q|^2

  const int tid = threadIdx.x;

  // Stage -2*W into padded LDS (coalesced: consecutive lanes, consecutive elems).
  for (int i = tid; i < NCODE * CDIM; i += 256) {
    int q = i >> 6, c = i & 63;
    ldsW[q * LDS_STRIDE + c] = -2.0f * W[i];
  }
  __syncthreads();
  // Per-code squared norms from staged values: w^2 = (v/-2)^2 = 0.25*v^2.
  for (int q = tid; q < NCODE; q += 256) {
    float s = 0.f;
    for (int c = 0; c < CDIM; ++c) { float v = ldsW[q * LDS_STRIDE + c]; s += v * v; }
    wnorm[q] = 0.25f * s;
  }
  __syncthreads();

  const int lane = tid & 31;
  const int wv   = tid >> 5;          // wave in block (8 wave32s)
  const int half = lane >> 4;         // 0: lanes 0-15, 1: lanes 16-31
  const int mrow = lane & 15;
  const int pair = blockIdx.x * 8 + wv;   // 256 blocks * 8 waves = 2048 pairs
  const int nb0  = pair * 32;             // two 16-pixel tiles per wave
  const int nb1  = nb0 + 16;

  // B fragments: X^T (K=dim, N=pixel). f32 4x16 B layout:
  //   VGPR r, lane l:  K = 4*s + r + 2*(l>>4),  N = l&15.
  v2f b0[16], b1[16];
  #pragma unroll
  for (int s = 0; s < 16; ++s) {
    int c  = 4 * s + 2 * half;
    int p0 = nb0 + mrow, p1 = nb1 + mrow;
    b0[s].x = x[xoff(p0, c)];  b0[s].y = x[xoff(p0, c + 1)];
    b1[s].x = x[xoff(p1, c)];  b1[s].y = x[xoff(p1, c + 1)];
  }

  float best0 = 3.4e38f, best1 = 3.4e38f;
  int   bi0 = 0, bi1 = 0;

  for (int qt = 0; qt < 64; ++qt) {       // 64 code tiles of 16 codes
    const int q0 = qt * 16;

    // Seed accumulator with |W_q|^2: acc VGPR r holds code q0 + r + 8*half.
    v8f accInit;
    #pragma unroll
    for (int r = 0; r < 8; ++r) accInit[r] = wnorm[q0 + r + 8 * half];
    v8f acc0v = accInit;
    v8f acc1v = accInit;

    // A fragments: -2*W tile (M=code, K=dim). f32 16x4 A layout:
    //   VGPR r, lane l:  M = l&15,  K = 4*s + r + 2*(l>>4).
    const float* arow = &ldsW[(q0 + mrow) * LDS_STRIDE + 2 * half];
    #pragma unroll
    for (int s = 0; s < 16; ++s) {        // C=64 -> 16 K=4 steps
      v2f a;
      a.x = arow[4 * s];
      a.y = arow[4 * s + 1];
      acc0v = __builtin_amdgcn_wmma_f32_16x16x4_f32(
          false, a, false, b0[s], (short)0, acc0v, false, false);
      acc1v = __builtin_amdgcn_wmma_f32_16x16x4_f32(
          false, a, false, b1[s], (short)0, acc1v, false, false);
    }

    // Per-lane running min; codes visited in strictly ascending order per lane,
    // so strict '<' implements argmin's first-index tie-break.
    #pragma unroll
    for (int r = 0; r < 8; ++r) {
      int code = q0 + r + 8 * half;
      float v0a = acc0v[r];
      if (v0a < best0) { best0 = v0a; bi0 = code; }
      float v1a = acc1v[r];
      if (v1a < best1) { best1 = v1a; bi1 = code; }
    }
  }

  // Merge halves: lane^16 holds the other 8 codes of the same pixel column.
  {
    float ov = __shfl_xor(best0, 16, 32);
    int   oi = __shfl_xor(bi0,   16, 32);
    if (ov < best0 || (ov == best0 && oi < bi0)) { best0 = ov; bi0 = oi; }
    ov = __shfl_xor(best1, 16, 32);
    oi = __shfl_xor(bi1,   16, 32);
    if (ov < best1 || (ov == best1 && oi < bi1)) { best1 = ov; bi1 = oi; }
  }
  if (lane < 16) {
    outIdx[nb0 + lane] = bi0;
    outIdx[nb1 + lane] = bi1;
  }
}

__global__ __launch_bounds__(256) void vq_gather(
    const float* __restrict__ x, const float* __restrict__ W,
    const int* __restrict__ idx, float* __restrict__ outQ,
    float* __restrict__ partial)
{
  __shared__ float red[256];
  const int g = blockIdx.x * 256 + threadIdx.x;   // 64*65536 elements
  const int c = g >> 16;
  const int n = g & (NPIX - 1);
  const int q = idx[n];                 // coalesced over n
  const float wvv = W[q * CDIM + c];    // gather; W is L2-resident (256KB)
  const int o = xoff(n, c);             // coalesced over n
  const float xv = x[o];
  outQ[o] = wvv;
  const float d = wvv - xv;
  red[threadIdx.x] = d * d;
  __syncthreads();
  for (int s2 = 128; s2 > 0; s2 >>= 1) {
    if (threadIdx.x < s2) red[threadIdx.x] += red[threadIdx.x + s2];
    __syncthreads();
  }
  if (threadIdx.x == 0) partial[blockIdx.x] = red[0];
}

__global__ __launch_bounds__(256) void vq_loss(
    const float* __restrict__ partial, float* __restrict__ out)
{
  __shared__ float red[256];
  float s = 0.f;
  for (int j = 0; j < 64; ++j) s += partial[threadIdx.x + 256 * j];  // fixed order
  red[threadIdx.x] = s;
  __syncthreads();
  for (int s2 = 128; s2 > 0; s2 >>= 1) {
    if (threadIdx.x < s2) red[threadIdx.x] += red[threadIdx.x + s2];
    __syncthreads();
  }
  if (threadIdx.x == 0) {
    const float loss = red[0] * (1.0f / 4194304.0f);
    out[4194304] = loss;   // quantization_loss
    out[4194305] = loss;   // commitment_loss (same value)
  }
}

extern "C" void kernel_launch(void* const* d_in, const int* in_sizes, int n_in,
                              void* d_out, int out_size, void* d_ws, size_t ws_size,
                              hipStream_t stream) {
  const float* x = (const float*)d_in[0];   // (16,64,64,64) f32
  const float* W = (const float*)d_in[1];   // (1024,64) f32
  float* out = (float*)d_out;               // [4194304 quantized | loss | loss]

  int*   idx     = (int*)d_ws;                                  // 65536 ints
  float* partial = (float*)((char*)d_ws + NPIX * sizeof(int));  // 16384 floats

  const size_t ldsBytes = (size_t)(NCODE * LDS_STRIDE + NCODE) * sizeof(float); // 270336

  vq_argmin_wmma<<<256, 256, ldsBytes, stream>>>(x, W, idx);
  vq_gather<<<(64 * NPIX) / 256, 256, 0, stream>>>(x, W, idx, out, partial);
  vq_loss<<<1, 256, 0, stream>>>(partial, out);
}